// CentralPathNet_26104811225809
// MI455X (gfx1250) — compile-verified
//
#include <hip/hip_runtime.h>
#include <math.h>
#include <stdint.h>

typedef __attribute__((ext_vector_type(2))) float v2f;
typedef __attribute__((ext_vector_type(8))) float v8f;

#define BATCH 32768
#define D_IN  512
#define H1D   256
#define H2D   256
#define H3D   128
#define ODIM  2112
#define N2    1024
#define MDIM  64
#define MIN_EIG_F 1e-4f

// d_out layout: [X_spd: BATCH*1024][y: BATCH*64][S_spd: BATCH*1024]
#define Y_OFS ((size_t)BATCH * N2)
#define S_OFS ((size_t)BATCH * (N2 + MDIM))

// ---------------------------------------------------------------------------
// Tiled GEMM with V_WMMA_F32_16X16X4_F32.
// Block: 256 threads = 8 waves.  Block tile 64(M) x 64(N); wave (wr,wc) owns a
// 16x32 strip (two independent WMMA accumulator chains for ILP).
// K is chunked by 32 through double-buffered LDS, filled with
// global_load_async_to_lds_b128 (ASYNCcnt) so the next chunk streams in while
// the current one is multiplied.  Compile-time K/NC/MODE -> constant strides.
// MODE 0: C = relu(A@W + b).   MODE 1: head scatter (+bh) into X / y / S.
// ---------------------------------------------------------------------------
template <int K, int NC, int MODE>
__global__ __launch_bounds__(256)
void gemm_wmma_f32_kernel(const float* __restrict__ A,
                          const float* __restrict__ W,
                          const float* __restrict__ bias,
                          float* __restrict__ Cout)
{
    __shared__ float As[2][64][36];   // row = 144 B (b128-aligned), conflict-free frags
    __shared__ float Bs[2][32][72];   // row = 288 B; k and k+2 rows 16 banks apart

    const int tid  = threadIdx.x;
    const int lane = tid & 31;
    const int wave = tid >> 5;
    const int wr   = wave & 3;                 // 4 row tiles of 16
    const int wc   = wave >> 2;                // 2 col strips of 32
    const int mb   = blockIdx.x * 64;
    const int nb   = blockIdx.y * 64;

    const int frow = lane & 15;                // M (A frag) / N (B frag)
    const int kq   = (lane >> 4) * 2;          // K sub-offset per ISA layout

    // Async global -> LDS staging, 16 B per lane per issue.
    // A tile: 64x32 = 512 float4 chunks;  B tile: 32x64 = 512 float4 chunks.
    auto stageA = [&](int buf, int kb, int c) {
        int r = c >> 3, col = (c & 7) * 4;
        unsigned long long g =
            (unsigned long long)(uintptr_t)(A + (size_t)(mb + r) * K + (kb + col));
        unsigned lds = (unsigned)(uintptr_t)&As[buf][r][col];
        asm volatile("global_load_async_to_lds_b128 %0, %1, off"
                     :: "v"(lds), "v"(g) : "memory");
    };
    auto stageB = [&](int buf, int kb, int c) {
        int r = c >> 4, col = (c & 15) * 4;
        unsigned long long g =
            (unsigned long long)(uintptr_t)(W + (size_t)(kb + r) * NC + (nb + col));
        unsigned lds = (unsigned)(uintptr_t)&Bs[buf][r][col];
        asm volatile("global_load_async_to_lds_b128 %0, %1, off"
                     :: "v"(lds), "v"(g) : "memory");
    };

    v8f acc0 = {0.f, 0.f, 0.f, 0.f, 0.f, 0.f, 0.f, 0.f};
    v8f acc1 = {0.f, 0.f, 0.f, 0.f, 0.f, 0.f, 0.f, 0.f};

    // Prefetch chunk 0.
    stageA(0, 0, tid); stageA(0, 0, tid + 256);
    stageB(0, 0, tid); stageB(0, 0, tid + 256);
    asm volatile("s_wait_asynccnt 0" ::: "memory");
    __syncthreads();

    for (int kb = 0; kb < K; kb += 32) {
        const int cur = (kb >> 5) & 1;
        const int nxt = cur ^ 1;
        if (kb + 32 < K) {                       // prefetch next chunk
            stageA(nxt, kb + 32, tid); stageA(nxt, kb + 32, tid + 256);
            stageB(nxt, kb + 32, tid); stageB(nxt, kb + 32, tid + 256);
        }

        // Pull all fragments for this chunk into registers (one DS wait),
        // then issue 16 WMMAs as two interleaved dependency chains.
        v2f a[8], b0[8], b1[8];
        #pragma unroll
        for (int t = 0; t < 8; ++t) {
            a[t].x  = As[cur][wr * 16 + frow][t * 4 + kq];
            a[t].y  = As[cur][wr * 16 + frow][t * 4 + kq + 1];
            b0[t].x = Bs[cur][t * 4 + kq][wc * 32 + frow];
            b0[t].y = Bs[cur][t * 4 + kq + 1][wc * 32 + frow];
            b1[t].x = Bs[cur][t * 4 + kq][wc * 32 + 16 + frow];
            b1[t].y = Bs[cur][t * 4 + kq + 1][wc * 32 + 16 + frow];
        }
        #pragma unroll
        for (int t = 0; t < 8; ++t) {
            acc0 = __builtin_amdgcn_wmma_f32_16x16x4_f32(
                       false, a[t], false, b0[t], (short)0, acc0, false, false);
            acc1 = __builtin_amdgcn_wmma_f32_16x16x4_f32(
                       false, a[t], false, b1[t], (short)0, acc1, false, false);
        }

        asm volatile("s_wait_asynccnt 0" ::: "memory");
        __syncthreads();                         // buffers swap safely
    }

    // Epilogue.  C/D layout: lane%16 = N, VGPR r -> M = r + 8*(lane/16).
    const int col0 = nb + wc * 32 + frow;
    const int col1 = col0 + 16;
    const float bv0 = bias[col0];
    const float bv1 = bias[col1];

    if (MODE == 0) {
        #pragma unroll
        for (int r = 0; r < 8; ++r) {
            int row = mb + wr * 16 + r + 8 * (lane >> 4);
            float u0 = acc0[r] + bv0, u1 = acc1[r] + bv1;
            Cout[(size_t)row * NC + col0] = u0 > 0.f ? u0 : 0.f;
            Cout[(size_t)row * NC + col1] = u1 > 0.f ? u1 : 0.f;
        }
    } else {
        // Head: region boundaries (1024, 1088) are 64-tile aligned -> uniform.
        float* dst; int cofs, rs;
        if (nb < N2)              { dst = Cout;         cofs = 0;         rs = N2;   }
        else if (nb < N2 + MDIM)  { dst = Cout + Y_OFS; cofs = N2;        rs = MDIM; }
        else                      { dst = Cout + S_OFS; cofs = N2 + MDIM; rs = N2;   }
        #pragma unroll
        for (int r = 0; r < 8; ++r) {
            int row = mb + wr * 16 + r + 8 * (lane >> 4);
            dst[(size_t)row * rs + (col0 - cofs)] = acc0[r] + bv0;
            dst[(size_t)row * rs + (col1 - cofs)] = acc1[r] + bv1;
        }
    }
}

// ---------------------------------------------------------------------------
// Batched symmetric 32x32 eigh via parallel cyclic Jacobi, one wave32 per
// matrix (65,536 workgroups).  Symmetrize -> 9 sweeps x 31 steps of 16
// disjoint Givens rotations (round-robin tournament pairing) -> clamp
// eigenvalues -> reconstruct V diag(w) V^T in place over d_out.
// Stride-33 LDS rows: conflict-free for both row- and column-indexed phases.
// ---------------------------------------------------------------------------
__global__ __launch_bounds__(32)
void jacobi_spd_kernel(float* __restrict__ out)
{
    __shared__ float Am[32][33];
    __shared__ float Vm[32][33];
    __shared__ float cs_c[16], cs_s[16];
    __shared__ int   cs_p[16], cs_q[16];
    __shared__ float wv[32];

    const int l   = threadIdx.x;
    const int mid = blockIdx.x;
    float* base = out + ((mid & 1) ? S_OFS : (size_t)0) + (size_t)(mid >> 1) * N2;

    #pragma unroll
    for (int i = 0; i < 32; ++i) {
        Am[i][l] = base[i * 32 + l];
        Vm[i][l] = (i == l) ? 1.f : 0.f;
    }
    __syncthreads();
    for (int i = 0; i < l; ++i) {               // symmetrize, pair (i,l) i<l
        float s = 0.5f * (Am[i][l] + Am[l][i]);
        Am[i][l] = s;
        Am[l][i] = s;
    }
    __syncthreads();

    for (int sweep = 0; sweep < 9; ++sweep) {
        for (int step = 0; step < 31; ++step) {
            if (l < 16) {                        // tournament pair schedule
                int p, q;
                if (l == 0) { p = 0;                       q = 1 + (30 + step) % 31; }
                else        { p = 1 + (l - 1 + step) % 31; q = 1 + (30 - l + step) % 31; }
                float app = Am[p][p], aqq = Am[q][q], apq = Am[p][q];
                float c = 1.f, s = 0.f;
                if (fabsf(apq) > 1e-20f) {
                    float tau = (aqq - app) / (2.f * apq);
                    float t = (tau >= 0.f ? 1.f : -1.f) /
                              (fabsf(tau) + sqrtf(1.f + tau * tau));
                    c = rsqrtf(1.f + t * t);
                    s = t * c;
                }
                cs_c[l] = c; cs_s[l] = s; cs_p[l] = p; cs_q[l] = q;
            }
            __syncthreads();
            #pragma unroll 4
            for (int r = 0; r < 16; ++r) {       // G^T A : lane = column
                int p = cs_p[r], q = cs_q[r];
                float c = cs_c[r], s = cs_s[r];
                float ap = Am[p][l], aq = Am[q][l];
                Am[p][l] = c * ap - s * aq;
                Am[q][l] = s * ap + c * aq;
            }
            __syncthreads();
            #pragma unroll 4
            for (int r = 0; r < 16; ++r) {       // A G and V G : lane = row
                int p = cs_p[r], q = cs_q[r];
                float c = cs_c[r], s = cs_s[r];
                float ap = Am[l][p], aq = Am[l][q];
                Am[l][p] = c * ap - s * aq;
                Am[l][q] = s * ap + c * aq;
                float vp = Vm[l][p], vq = Vm[l][q];
                Vm[l][p] = c * vp - s * vq;
                Vm[l][q] = s * vp + c * vq;
            }
            __syncthreads();
        }
    }

    wv[l] = fmaxf(Am[l][l], MIN_EIG_F);
    __syncthreads();
    float t[32];
    #pragma unroll
    for (int k = 0; k < 32; ++k) t[k] = wv[k] * Vm[l][k];    // lane l = col j
    #pragma unroll 4
    for (int i = 0; i < 32; ++i) {
        float acc = 0.f;
        #pragma unroll
        for (int k = 0; k < 32; ++k) acc += Vm[i][k] * t[k]; // broadcast reads
        base[i * 32 + l] = acc;                              // coalesced store
    }
}

// ---------------------------------------------------------------------------
extern "C" void kernel_launch(void* const* d_in, const int* in_sizes, int n_in,
                              void* d_out, int out_size, void* d_ws, size_t ws_size,
                              hipStream_t stream) {
    const float* x  = (const float*)d_in[0];
    const float* W1 = (const float*)d_in[1];
    const float* b1 = (const float*)d_in[2];
    const float* W2 = (const float*)d_in[3];
    const float* b2 = (const float*)d_in[4];
    const float* W3 = (const float*)d_in[5];
    const float* b3 = (const float*)d_in[6];
    const float* Wh = (const float*)d_in[7];
    const float* bh = (const float*)d_in[8];
    float* out = (float*)d_out;

    float* h1 = (float*)d_ws;                    // 32 MB
    float* h2 = h1 + (size_t)BATCH * H1D;        // 32 MB
    float* h3 = h2 + (size_t)BATCH * H2D;        // 16 MB

    dim3 blk(256);
    gemm_wmma_f32_kernel<D_IN, H1D, 0>
        <<<dim3(BATCH / 64, H1D / 64), blk, 0, stream>>>(x,  W1, b1, h1);
    gemm_wmma_f32_kernel<H1D, H2D, 0>
        <<<dim3(BATCH / 64, H2D / 64), blk, 0, stream>>>(h1, W2, b2, h2);
    gemm_wmma_f32_kernel<H2D, H3D, 0>
        <<<dim3(BATCH / 64, H3D / 64), blk, 0, stream>>>(h2, W3, b3, h3);
    gemm_wmma_f32_kernel<H3D, ODIM, 1>
        <<<dim3(BATCH / 64, ODIM / 64), blk, 0, stream>>>(h3, Wh, bh, out);
    jacobi_spd_kernel<<<dim3(2 * BATCH), dim3(32), 0, stream>>>(out);
}